// SAGEBlock_28312424415601
// MI455X (gfx1250) — compile-verified
//
#include <hip/hip_runtime.h>
#include <hip/hip_bf16.h>
#include <math.h>

typedef __attribute__((ext_vector_type(2))) float v2f;
typedef __attribute__((ext_vector_type(8))) float v8f;

// ---------------------------------------------------------------------------
// Kernel 0: zero the workspace (agg_sum[N*64] + count[N])
// ---------------------------------------------------------------------------
__global__ void sage_zero_kernel(float* __restrict__ p, size_t n) {
    size_t i = (size_t)blockIdx.x * blockDim.x + threadIdx.x;
    size_t stride = (size_t)gridDim.x * blockDim.x;
    for (; i < n; i += stride) p[i] = 0.0f;
}

// ---------------------------------------------------------------------------
// Kernel 1: edge scatter.  16 threads per edge, float4 per thread.
//   agg_sum[dst] += x[src];  count[dst] += 1
// x (25.6MB) and agg_sum (25.6MB) both fit in the 192MB L2, so the random
// gather/atomic traffic stays on-die; HBM only sees streaming reads of the
// edge list.
// ---------------------------------------------------------------------------
__global__ __launch_bounds__(256) void sage_scatter_kernel(
    const float* __restrict__ x,
    const long long* __restrict__ ei,   // [2, E] int64
    float* __restrict__ aggsum,
    float* __restrict__ cnt,
    int E)
{
    int tid   = blockIdx.x * blockDim.x + threadIdx.x;
    int e     = tid >> 4;
    int chunk = tid & 15;
    if (e >= E) return;
    long long src = ei[e];
    long long dst = ei[(size_t)E + e];
    const float4 v = *(const float4*)(x + (size_t)src * 64 + chunk * 4);
    float* a = aggsum + (size_t)dst * 64 + chunk * 4;
    atomicAdd(a + 0, v.x);
    atomicAdd(a + 1, v.y);
    atomicAdd(a + 2, v.z);
    atomicAdd(a + 3, v.w);
    if (chunk == 0) atomicAdd(cnt + dst, 1.0f);
}

// ---------------------------------------------------------------------------
// Kernel 2: fused  out = L2norm(relu((agg/deg)@Wl^T + b + x@Wr^T))
// One wave32 per 16-node tile, full 16x64 output via V_WMMA_F32_16X16X4_F32.
//   A frag (16x4 f32): lanes 0-15 -> M=lane, K=k0,k0+1 ; lanes 16-31 -> K=k0+2,k0+3
//   B frag (4x16 f32): lanes 0-15 -> N=lane, same K split
//   C/D   (16x16 f32): vgpr i, lanes 0-15 -> M=i ; lanes 16-31 -> M=i+8
// ---------------------------------------------------------------------------
__global__ __launch_bounds__(64) void sage_gemm_kernel(
    const float* __restrict__ x,
    const float* __restrict__ aggsum,
    const float* __restrict__ cnt,
    const float* __restrict__ Wl,   // [64,64] row-major [o][k]
    const float* __restrict__ Wr,   // [64,64] row-major [o][k]
    const float* __restrict__ bl,   // [64]
    float* __restrict__ out,        // [N,64]
    int N)
{
    const int wave = threadIdx.x >> 5;
    const int lane = threadIdx.x & 31;
    const int half = lane >> 4;       // which K-pair / which M-half
    const int l15  = lane & 15;

    const int tile = blockIdx.x * 2 + wave;
    const int base = tile * 16;
    if (base >= N) return;            // whole wave exits: EXEC all-1s at WMMA

    // A-fragment row for this lane is base + (lane & 15) for BOTH halves.
    int rowA = base + l15;
    if (rowA >= N) rowA = N - 1;      // clamp (N%16==0 here, pure safety)

    const float invc = 1.0f / fmaxf(cnt[rowA], 1.0f);
    const float* __restrict__ xrow = x      + (size_t)rowA * 64;
    const float* __restrict__ arow = aggsum + (size_t)rowA * 64;

    v8f acc0 = {}, acc1 = {}, acc2 = {}, acc3 = {};
    v8f* acc[4] = { &acc0, &acc1, &acc2, &acc3 };

    #pragma unroll
    for (int k0 = 0; k0 < 64; k0 += 4) {
        const int ka = k0 + 2 * half;
        v2f a_self, a_agg;
        a_self.x = xrow[ka];
        a_self.y = xrow[ka + 1];
        a_agg.x  = arow[ka]     * invc;
        a_agg.y  = arow[ka + 1] * invc;

        #pragma unroll
        for (int t = 0; t < 4; ++t) {
            const int col = l15 + 16 * t;   // output channel for this lane
            v2f bfl, bfr;
            bfl.x = Wl[col * 64 + ka];
            bfl.y = Wl[col * 64 + ka + 1];
            bfr.x = Wr[col * 64 + ka];
            bfr.y = Wr[col * 64 + ka + 1];
            // D = A*B + C   (neg_a, A, neg_b, B, c_mod, C, reuse_a, reuse_b)
            *acc[t] = __builtin_amdgcn_wmma_f32_16x16x4_f32(
                false, a_agg,  false, bfl, (short)0, *acc[t], false, false);
            *acc[t] = __builtin_amdgcn_wmma_f32_16x16x4_f32(
                false, a_self, false, bfr, (short)0, *acc[t], false, false);
        }
    }

    // bias + relu, accumulate per-lane partial sum-of-squares per C row
    float bias[4];
    #pragma unroll
    for (int t = 0; t < 4; ++t) bias[t] = bl[l15 + 16 * t];

    float p[8];
    #pragma unroll
    for (int i = 0; i < 8; ++i) {
        float s = 0.0f;
        #pragma unroll
        for (int t = 0; t < 4; ++t) {
            float v = (*acc[t])[i] + bias[t];
            v = fmaxf(v, 0.0f);
            (*acc[t])[i] = v;
            s += v * v;
        }
        p[i] = s;
    }

    // Row sum-of-squares: reduce across the 16 lanes of each half-wave
    // (masks 1/2/4/8 never cross the half boundary — matches C-row layout).
    #pragma unroll
    for (int i = 0; i < 8; ++i) {
        float s = p[i];
        s += __shfl_xor(s, 1, 32);
        s += __shfl_xor(s, 2, 32);
        s += __shfl_xor(s, 4, 32);
        s += __shfl_xor(s, 8, 32);
        p[i] = 1.0f / (sqrtf(s) + 1e-9f);
    }

    // Scaled coalesced stores: row = base + i + 8*half, col = l15 + 16t
    #pragma unroll
    for (int i = 0; i < 8; ++i) {
        const int row = base + i + 8 * half;
        if (row >= N) continue;
        float* __restrict__ orow = out + (size_t)row * 64;
        #pragma unroll
        for (int t = 0; t < 4; ++t)
            orow[l15 + 16 * t] = (*acc[t])[i] * p[i];
    }
}

// ---------------------------------------------------------------------------
extern "C" void kernel_launch(void* const* d_in, const int* in_sizes, int n_in,
                              void* d_out, int out_size, void* d_ws, size_t ws_size,
                              hipStream_t stream) {
    const float*     x  = (const float*)d_in[0];      // [N,64] f32
    const long long* ei = (const long long*)d_in[1];  // [2,E] i64
    const float*     Wl = (const float*)d_in[2];      // [64,64]
    const float*     bl = (const float*)d_in[3];      // [64]
    const float*     Wr = (const float*)d_in[4];      // [64,64]
    float*           out = (float*)d_out;

    const int N = in_sizes[0] / 64;
    const int E = in_sizes[1] / 2;

    float* aggsum = (float*)d_ws;                 // N*64 floats
    float* cntbuf = aggsum + (size_t)N * 64;      // N floats

    // 0) zero workspace
    const size_t zn = (size_t)N * 64 + (size_t)N;
    int zblocks = (int)((zn + 255) / 256);
    if (zblocks > 16384) zblocks = 16384;
    sage_zero_kernel<<<zblocks, 256, 0, stream>>>(aggsum, zn);

    // 1) scatter-add (16 threads per edge)
    const long long sthreads = (long long)E * 16;
    const int sblocks = (int)((sthreads + 255) / 256);
    sage_scatter_kernel<<<sblocks, 256, 0, stream>>>(x, ei, aggsum, cntbuf, E);

    // 2) fused mean-div + dual WMMA GEMM + bias + relu + L2 norm
    const int tiles  = (N + 15) / 16;
    const int gblocks = (tiles + 1) / 2;          // 2 waves (tiles) per block
    sage_gemm_kernel<<<gblocks, 64, 0, stream>>>(x, aggsum, cntbuf,
                                                 Wl, Wr, bl, out, N);
}